// TasteGNN_16432544874506
// MI455X (gfx1250) — compile-verified
//
#include <hip/hip_runtime.h>
#include <math.h>

// ---------------------------------------------------------------------------
// HANConv-style taste GNN for MI455X (gfx1250, wave32).
//  - hs GEMM (x_ing @ W_ing^T + b) via V_WMMA_F32_16X16X4_F32 (fp32, exact).
//  - hd GEMM eliminated: a_dst = x_taste @ (W_taste^T att_dst) + b.att_dst.
//  - semantic attention eliminated: softmax over a single meta-path => beta==1.
//  - segment softmax via order-preserving uint atomicMax + float atomicAdd.
//  - residual + batch-stat BatchNorm + ReLU fused, accumulator lives in d_out.
// ---------------------------------------------------------------------------

#define HID     128
#define N_ING   20000
#define N_TASTE 100000
#define N_EDGE  600000

typedef __attribute__((ext_vector_type(2))) float v2f;
typedef __attribute__((ext_vector_type(8))) float v8f;

// order-preserving float<->uint encoding for atomicMax on floats
__device__ __forceinline__ unsigned enc_f(float f) {
  unsigned u = __float_as_uint(f);
  return (u & 0x80000000u) ? ~u : (u | 0x80000000u);
}
__device__ __forceinline__ float dec_f(unsigned u) {
  return (u & 0x80000000u) ? __uint_as_float(u & 0x7FFFFFFFu)
                           : __uint_as_float(~u);
}

// ---------------------------------------------------------------------------
// hs = x_ing @ W^T + b   via WMMA f32 16x16x4.
// block = (32, 8): 8 waves, each wave owns one 16x16 output tile
// (wave y covers output cols [16y, 16y+16)); blockIdx.x = 16-row tile.
// A 16x4 tile: lane<16 -> K=k0+{0,1}; lane>=16 -> K=k0+{2,3}; row = lane&15.
// B 4x16 tile mirrored; col = lane&15.  D: VGPR v -> row v + 8*(lane>=16).
// ---------------------------------------------------------------------------
__global__ __launch_bounds__(256) void k_gemm_hs(
    const float* __restrict__ X, const float* __restrict__ W,
    const float* __restrict__ bias, float* __restrict__ hs) {
  const int lane = threadIdx.x;
  const int half = lane >> 4;
  const int l    = lane & 15;
  const int row0 = blockIdx.x * 16;
  const int col0 = threadIdx.y * 16;
  const int arow = row0 + l;
  const int bcol = col0 + l;

  const float2* Xp = (const float2*)(X + (size_t)arow * HID + half * 2);
  const float2* Wp = (const float2*)(W + (size_t)bcol * HID + half * 2);

  v8f acc = {};
#pragma unroll 4
  for (int k0 = 0; k0 < HID; k0 += 4) {
    float2 af = Xp[k0 >> 1];           // X[arow, k0 + 2*half + {0,1}]
    float2 bf = Wp[k0 >> 1];           // W[bcol, k0 + 2*half + {0,1}]
    v2f a; a.x = af.x; a.y = af.y;
    v2f b; b.x = bf.x; b.y = bf.y;
    acc = __builtin_amdgcn_wmma_f32_16x16x4_f32(
        /*neg_a=*/false, a, /*neg_b=*/false, b,
        /*c_mod=*/(short)0, acc, /*reuse_a=*/false, /*reuse_b=*/false);
  }
  const float bv = bias[col0 + l];
#pragma unroll
  for (int v = 0; v < 8; ++v) {
    int m = v + half * 8;
    hs[(size_t)(row0 + m) * HID + col0 + l] = acc[v] + bv;
  }
}

// one wave per row: out[row] = dot(M[row,:], vvec) (+ *cadd if given)
__global__ __launch_bounds__(256) void k_rowdot(
    const float* __restrict__ M, const float* __restrict__ vvec,
    const float* __restrict__ cadd, float* __restrict__ out, int nrows) {
  int wave = blockIdx.x * (blockDim.x >> 5) + (threadIdx.x >> 5);
  int lane = threadIdx.x & 31;
  if (wave >= nrows) return;
  const float* row = M + (size_t)wave * HID;
  float s = 0.f;
#pragma unroll
  for (int j = 0; j < HID / 32; ++j)
    s += row[lane + 32 * j] * vvec[lane + 32 * j];
#pragma unroll
  for (int off = 16; off > 0; off >>= 1) s += __shfl_xor(s, off, 32);
  if (lane == 0) out[wave] = s + (cadd ? *cadd : 0.f);
}

// u[k] = sum_d W[d,k]*att[d];  *cconst = dot(b, att)
__global__ void k_uvec(const float* __restrict__ W,
                       const float* __restrict__ bvec,
                       const float* __restrict__ att,
                       float* __restrict__ u, float* __restrict__ cconst) {
  int k = threadIdx.x;
  float s = 0.f;
  for (int d = 0; d < HID; ++d) s += W[(size_t)d * HID + k] * att[d];
  u[k] = s;
  if (k == 0) {
    float c = 0.f;
    for (int d = 0; d < HID; ++d) c += bvec[d] * att[d];
    *cconst = c;
  }
}

__global__ void k_init(float* __restrict__ outacc, float* __restrict__ denom,
                       unsigned* __restrict__ amax, float* __restrict__ stats) {
  size_t i = (size_t)blockIdx.x * blockDim.x + threadIdx.x;
  const size_t total = (size_t)N_TASTE * HID;
  if (i < total) outacc[i] = 0.f;
  if (i < N_TASTE) { denom[i] = 0.f; amax[i] = 0u; }  // 0u == encoded -min
  if (i < 2 * HID) stats[i] = 0.f;
}

// pass 1: alpha = leaky_relu(a_src[s]+a_dst[d], 0.2); segment max via atomicMax
__global__ void k_edge1(const int* __restrict__ src, const int* __restrict__ dst,
                        const float* __restrict__ a_src,
                        const float* __restrict__ a_dst,
                        float* __restrict__ alpha, unsigned* __restrict__ amax) {
  int e = blockIdx.x * blockDim.x + threadIdx.x;
  if (e >= N_EDGE) return;
  int d = dst[e];
  float a = a_src[src[e]] + a_dst[d];
  a = (a >= 0.f) ? a : 0.2f * a;
  alpha[e] = a;
  atomicMax(&amax[d], enc_f(a));
}

// pass 2: e = exp(alpha - amax[dst]); denom[dst] += e
__global__ void k_edge2(const int* __restrict__ dst, float* __restrict__ alpha,
                        const unsigned* __restrict__ amax,
                        float* __restrict__ denom) {
  int e = blockIdx.x * blockDim.x + threadIdx.x;
  if (e >= N_EDGE) return;
  int d = dst[e];
  float ex = expf(alpha[e] - dec_f(amax[d]));
  alpha[e] = ex;
  atomicAdd(&denom[d], ex);
}

// pass 3: one wave per edge, outacc[dst,:] += hs[src,:] * w (coalesced atomics)
__global__ __launch_bounds__(256) void k_edge3(
    const int* __restrict__ src, const int* __restrict__ dst,
    const float* __restrict__ hs, const float* __restrict__ alpha,
    const float* __restrict__ denom, float* __restrict__ outacc) {
  int wave = blockIdx.x * (blockDim.x >> 5) + (threadIdx.x >> 5);
  int lane = threadIdx.x & 31;
  if (wave >= N_EDGE) return;
  int s = src[wave], d = dst[wave];
  float w = alpha[wave] / (denom[d] + 1e-16f);
  const float* hrow = hs + (size_t)s * HID;
  float* orow = outacc + (size_t)d * HID;
#pragma unroll
  for (int j = 0; j < HID / 32; ++j)
    atomicAdd(&orow[lane + 32 * j], hrow[lane + 32 * j] * w);
}

// relu(msg) + residual; accumulate per-channel batch stats (thread = channel)
__global__ __launch_bounds__(128) void k_fin1(
    const float* __restrict__ x_taste, float* __restrict__ t,
    float* __restrict__ gsum, float* __restrict__ gsumsq, int rows_per_block) {
  int c  = threadIdx.x;
  int r0 = blockIdx.x * rows_per_block;
  float s = 0.f, s2 = 0.f;
  for (int r = 0; r < rows_per_block; ++r) {
    int row = r0 + r;
    if (row >= N_TASTE) break;
    size_t idx = (size_t)row * HID + c;
    float v = fmaxf(t[idx], 0.f) + x_taste[idx];
    t[idx] = v;
    s += v; s2 += v * v;
  }
  atomicAdd(&gsum[c], s);
  atomicAdd(&gsumsq[c], s2);
}

__global__ void k_bnp(const float* __restrict__ gsum,
                      const float* __restrict__ gsumsq,
                      const float* __restrict__ gamma,
                      const float* __restrict__ beta,
                      float* __restrict__ scale, float* __restrict__ shift) {
  int c = threadIdx.x;
  float inv_n = 1.f / (float)N_TASTE;
  float mean = gsum[c] * inv_n;
  float var  = gsumsq[c] * inv_n - mean * mean;   // population var (jnp.var)
  float sc   = gamma[c] * rsqrtf(var + 1e-5f);
  scale[c] = sc;
  shift[c] = beta[c] - mean * sc;
}

__global__ void k_fin2(float* __restrict__ t, const float* __restrict__ scale,
                       const float* __restrict__ shift) {
  size_t i = (size_t)blockIdx.x * blockDim.x + threadIdx.x;
  if (i >= (size_t)N_TASTE * HID) return;
  int c = (int)(i & (HID - 1));
  t[i] = fmaxf(t[i] * scale[c] + shift[c], 0.f);
}

__global__ void k_copy4(const float4* __restrict__ src, float4* __restrict__ dst,
                        int n4) {
  int i = blockIdx.x * blockDim.x + threadIdx.x;
  if (i < n4) dst[i] = src[i];
}

// ---------------------------------------------------------------------------
extern "C" void kernel_launch(void* const* d_in, const int* in_sizes, int n_in,
                              void* d_out, int out_size, void* d_ws,
                              size_t ws_size, hipStream_t stream) {
  (void)in_sizes; (void)n_in; (void)out_size; (void)ws_size;

  const float* x_ing   = (const float*)d_in[0];
  const float* x_taste = (const float*)d_in[1];
  const int*   esrc    = (const int*)d_in[2];
  const int*   edst    = (const int*)d_in[3];
  const float* W_ing   = (const float*)d_in[4];
  const float* b_ing   = (const float*)d_in[5];
  const float* W_taste = (const float*)d_in[6];
  const float* b_taste = (const float*)d_in[7];
  const float* att_src = (const float*)d_in[8];
  const float* att_dst = (const float*)d_in[9];
  // d_in[10..12] (k_lin_w, k_lin_b, q_sem) unused: softmax over ONE meta-path
  // score is identically 1.0, so the semantic-attention block is a no-op.
  const float* bn_gamma = (const float*)d_in[13];
  const float* bn_beta  = (const float*)d_in[14];

  float* out       = (float*)d_out;
  float* taste_out = out + (size_t)N_ING * HID;  // also the msg accumulator

  // workspace carve-up (~14 MB)
  float* f      = (float*)d_ws;
  float* hs     = f;  f += (size_t)N_ING * HID;   // 2,560,000
  float* a_src  = f;  f += N_ING;                 // 20,000
  float* a_dst  = f;  f += N_TASTE;               // 100,000
  float* u_vec  = f;  f += HID;                   // 128
  float* c_scal = f;  f += 32;                    // 1 (padded)
  float* denom  = f;  f += N_TASTE;               // 100,000
  float* alpha  = f;  f += N_EDGE;                // 600,000
  float* gsum   = f;  f += HID;                   // 128
  float* gsumsq = f;  f += HID;                   // 128
  float* scale  = f;  f += HID;                   // 128
  float* shift  = f;  f += HID;                   // 128
  unsigned* amax = (unsigned*)f;                  // 100,000 uints

  // 0) zero accumulators / init segment-max
  {
    size_t total = (size_t)N_TASTE * HID;
    int nb = (int)((total + 255) / 256);
    k_init<<<nb, 256, 0, stream>>>(taste_out, denom, amax, gsum);
  }
  // 1) hs = x_ing @ W_ing^T + b  (WMMA fp32)
  k_gemm_hs<<<dim3(N_ING / 16), dim3(32, 8), 0, stream>>>(x_ing, W_ing, b_ing, hs);
  // 2) a_src = hs . att_src
  k_rowdot<<<N_ING / 8, 256, 0, stream>>>(hs, att_src, nullptr, a_src, N_ING);
  // 3) u = W_taste^T att_dst ; c = b_taste . att_dst   (kills the 100k GEMM)
  k_uvec<<<1, HID, 0, stream>>>(W_taste, b_taste, att_dst, u_vec, c_scal);
  // 4) a_dst = x_taste . u + c
  k_rowdot<<<N_TASTE / 8, 256, 0, stream>>>(x_taste, u_vec, c_scal, a_dst, N_TASTE);
  // 5) segment softmax (max, exp+sum, weighted scatter)
  k_edge1<<<(N_EDGE + 255) / 256, 256, 0, stream>>>(esrc, edst, a_src, a_dst, alpha, amax);
  k_edge2<<<(N_EDGE + 255) / 256, 256, 0, stream>>>(edst, alpha, amax, denom);
  k_edge3<<<N_EDGE / 8, 256, 0, stream>>>(esrc, edst, hs, alpha, denom, taste_out);
  // 6) relu + residual + batch stats
  k_fin1<<<(N_TASTE + 127) / 128, 128, 0, stream>>>(x_taste, taste_out, gsum, gsumsq, 128);
  // 7) BN affine params, then normalize + relu in place
  k_bnp<<<1, HID, 0, stream>>>(gsum, gsumsq, bn_gamma, bn_beta, scale, shift);
  {
    size_t total = (size_t)N_TASTE * HID;
    k_fin2<<<(int)((total + 255) / 256), 256, 0, stream>>>(taste_out, scale, shift);
  }
  // 8) passthrough ingredient output
  k_copy4<<<(N_ING * HID / 4 + 255) / 256, 256, 0, stream>>>(
      (const float4*)x_ing, (float4*)out, N_ING * HID / 4);
}